// GCN_68066641707903
// MI455X (gfx1250) — compile-verified
//
#include <hip/hip_runtime.h>
#include <hip/hip_bf16.h>

typedef __attribute__((ext_vector_type(2))) float v2f;
typedef __attribute__((ext_vector_type(8))) float v8f;

// ---------------- degree / dinv ----------------
__global__ void k_fill1(float* __restrict__ p, int n) {
    int i = blockIdx.x * blockDim.x + threadIdx.x;
    if (i < n) p[i] = 1.0f;   // self-loop contributes 1 to every node's degree
}

__global__ void k_deg(const int* __restrict__ dst, float* __restrict__ deg, int E) {
    int e = blockIdx.x * blockDim.x + threadIdx.x;
    if (e < E) atomicAdd(&deg[dst[e]], 1.0f);
}

__global__ void k_rsqrt_inplace(float* __restrict__ p, int n) {
    int i = blockIdx.x * blockDim.x + threadIdx.x;
    if (i < n) {
        float d = p[i];
        p[i] = (d > 0.0f) ? rsqrtf(d) : 0.0f;
    }
}

// ---------------- WMMA GEMM: out[M,64] = A[M,K] @ W[K,64] ----------------
// One wave computes a 16-row x 64-col strip. M % 16 == 0, K % 4 == 0.
__global__ void k_gemm_wmma(const float* __restrict__ A, const float* __restrict__ W,
                            float* __restrict__ out, int M, int K) {
    int wave = (blockIdx.x * blockDim.x + threadIdx.x) >> 5;
    int lane = threadIdx.x & 31;
    int mtiles = M >> 4;
    if (wave >= mtiles) return;          // wave-uniform exit; surviving waves have full EXEC

    const int half = lane >> 4;          // 0 or 1
    const int mr   = lane & 15;
    const int row0 = wave << 4;

    v8f acc[4] = {};                     // 4 N-tiles of 16x16 f32

    const float* __restrict__ arow = A + (size_t)(row0 + mr) * K;
    for (int kb = 0; kb < K; kb += 4) {
        const int ka = kb + 2 * half;
        v2f a;
        a.x = arow[ka];
        a.y = arow[ka + 1];
#pragma unroll
        for (int nt = 0; nt < 4; ++nt) {
            const int n = (nt << 4) + mr;
            v2f b;
            b.x = W[(size_t)ka * 64 + n];
            b.y = W[(size_t)(ka + 1) * 64 + n];
            acc[nt] = __builtin_amdgcn_wmma_f32_16x16x4_f32(
                false, a, false, b, (short)0, acc[nt], false, false);
        }
    }
#pragma unroll
    for (int nt = 0; nt < 4; ++nt) {
#pragma unroll
        for (int v = 0; v < 8; ++v) {
            const int m = row0 + v + 8 * half;
            const int n = (nt << 4) + mr;
            out[(size_t)m * 64 + n] = acc[nt][v];
        }
    }
}

// ---------------- aggregation ----------------
// agg[i,f] = xw[i,f] * dinv[i]^2   (self-loop term, also zero-initializes agg)
__global__ void k_selfloop_init(const float* __restrict__ xw, const float* __restrict__ dinv,
                                float* __restrict__ agg, int N) {
    int t = blockIdx.x * blockDim.x + threadIdx.x;
    if (t < N * 64) {
        int i = t >> 6;
        float di = dinv[i];
        agg[t] = xw[t] * di * di;
    }
}

// agg[dst,f] += xw[src,f] * dinv[src] * dinv[dst]   (64 threads per edge)
__global__ void k_edge_scatter(const int* __restrict__ src, const int* __restrict__ dst,
                               const float* __restrict__ xw, const float* __restrict__ dinv,
                               float* __restrict__ agg, int E) {
    long long t = (long long)blockIdx.x * blockDim.x + threadIdx.x;
    long long e = t >> 6;
    int f = (int)(t & 63);
    if (e < E) {
        int s = src[e], d = dst[e];
        float norm = dinv[s] * dinv[d];
        atomicAdd(&agg[(size_t)d * 64 + f], xw[(size_t)s * 64 + f] * norm);
    }
}

// h[i,f] = relu(agg[i,f] + b[f])   in place
__global__ void k_bias_relu(float* __restrict__ h, const float* __restrict__ b, int N) {
    int t = blockIdx.x * blockDim.x + threadIdx.x;
    if (t < N * 64) {
        float v = h[t] + b[t & 63];
        h[t] = v > 0.0f ? v : 0.0f;
    }
}

// ---------------- pooling ----------------
__global__ void k_zero(float* __restrict__ p, int n) {
    int i = blockIdx.x * blockDim.x + threadIdx.x;
    if (i < n) p[i] = 0.0f;
}

__global__ void k_pool(const float* __restrict__ h, const int* __restrict__ batch,
                       float* __restrict__ pooled, float* __restrict__ cnt, int N) {
    int t = blockIdx.x * blockDim.x + threadIdx.x;
    if (t < N * 64) {
        int i = t >> 6;
        int f = t & 63;
        int g = batch[i];
        atomicAdd(&pooled[(size_t)g * 64 + f], h[t]);
        if (f == 0) atomicAdd(&cnt[g], 1.0f);
    }
}

// ---------------- final FC + log_softmax + y passthrough ----------------
__global__ void k_final(const float* __restrict__ pooled, const float* __restrict__ cnt,
                        const float* __restrict__ Wfc, const float* __restrict__ bfc,
                        const int* __restrict__ y, float* __restrict__ out, int G) {
    int g = blockIdx.x * blockDim.x + threadIdx.x;
    if (g >= G) return;
    float denom = fmaxf(cnt[g], 1.0f);
    float inv = 1.0f / denom;
    float l0 = bfc[0], l1 = bfc[1], l2 = bfc[2], l3 = bfc[3];
    const float* pr = pooled + (size_t)g * 64;
    for (int k = 0; k < 64; ++k) {
        float pv = pr[k] * inv;
        const float* wr = Wfc + (size_t)k * 4;
        l0 = fmaf(pv, wr[0], l0);
        l1 = fmaf(pv, wr[1], l1);
        l2 = fmaf(pv, wr[2], l2);
        l3 = fmaf(pv, wr[3], l3);
    }
    float m = fmaxf(fmaxf(l0, l1), fmaxf(l2, l3));
    float s = expf(l0 - m) + expf(l1 - m) + expf(l2 - m) + expf(l3 - m);
    float lse = m + logf(s);
    float* o = out + (size_t)g * 4;
    o[0] = l0 - lse;
    o[1] = l1 - lse;
    o[2] = l2 - lse;
    o[3] = l3 - lse;
    out[(size_t)G * 4 + g] = (float)y[g];   // tuple's second element, cast to float
}

extern "C" void kernel_launch(void* const* d_in, const int* in_sizes, int n_in,
                              void* d_out, int out_size, void* d_ws, size_t ws_size,
                              hipStream_t stream) {
    const float* x      = (const float*)d_in[0];   // [N,128]
    const int*   ei     = (const int*)d_in[1];     // [2,E]
    const int*   batch  = (const int*)d_in[2];     // [N]
    const int*   y      = (const int*)d_in[3];     // [G]
    const float* W1     = (const float*)d_in[4];   // [128,64]
    const float* b1     = (const float*)d_in[5];
    const float* W2     = (const float*)d_in[6];   // [64,64]
    const float* b2     = (const float*)d_in[7];
    const float* Wfc    = (const float*)d_in[8];   // [64,4]
    const float* bfc    = (const float*)d_in[9];

    const int N = in_sizes[0] / 128;
    const int E = in_sizes[1] / 2;
    const int G = in_sizes[3];
    const int* src = ei;
    const int* dst = ei + E;

    float* ws = (float*)d_ws;
    size_t off = 0;
    float* dinv   = ws + off; off += ((size_t)N + 255) & ~(size_t)255;
    float* bufA   = ws + off; off += (size_t)N * 64;   // xw of current layer
    float* bufB   = ws + off; off += (size_t)N * 64;   // aggregated / activated
    float* pooled = ws + off; off += (size_t)G * 64;
    float* cnt    = ws + off; off += (size_t)G;
    (void)ws_size;

    const int TB = 256;
    const int nNode   = (N + TB - 1) / TB;
    const int nEdge   = (E + TB - 1) / TB;
    const int nNF     = ((N * 64) + TB - 1) / TB;
    const long long enf = (long long)E * 64;
    const int nEF     = (int)((enf + TB - 1) / TB);
    const int nGemm   = ((N / 16) + 7) / 8;            // 8 waves/block, 1 wave per 16-row strip
    const int nPool   = ((G * 64 + G) + TB - 1) / TB;
    const int nG      = (G + TB - 1) / TB;

    // degrees with self-loops -> dinv = rsqrt(deg)
    k_fill1<<<nNode, TB, 0, stream>>>(dinv, N);
    k_deg<<<nEdge, TB, 0, stream>>>(dst, dinv, E);
    k_rsqrt_inplace<<<nNode, TB, 0, stream>>>(dinv, N);

    // ---- layer 1 ----
    k_gemm_wmma<<<nGemm, TB, 0, stream>>>(x, W1, bufA, N, 128);
    k_selfloop_init<<<nNF, TB, 0, stream>>>(bufA, dinv, bufB, N);
    k_edge_scatter<<<nEF, TB, 0, stream>>>(src, dst, bufA, dinv, bufB, E);
    k_bias_relu<<<nNF, TB, 0, stream>>>(bufB, b1, N);

    // ---- layer 2 ----
    k_gemm_wmma<<<nGemm, TB, 0, stream>>>(bufB, W2, bufA, N, 64);
    k_selfloop_init<<<nNF, TB, 0, stream>>>(bufA, dinv, bufB, N);
    k_edge_scatter<<<nEF, TB, 0, stream>>>(src, dst, bufA, dinv, bufB, E);
    k_bias_relu<<<nNF, TB, 0, stream>>>(bufB, b2, N);

    // ---- mean pool + FC + log_softmax ----
    k_zero<<<nPool, TB, 0, stream>>>(pooled, G * 64 + G);   // pooled and cnt are contiguous
    k_pool<<<nNF, TB, 0, stream>>>(bufB, batch, pooled, cnt, N);
    k_final<<<nG, TB, 0, stream>>>(pooled, cnt, Wfc, bfc, y, (float*)d_out, G);
}